// DCTLinearFactored_44014824849786
// MI455X (gfx1250) — compile-verified
//
#include <hip/hip_runtime.h>
#include <hip/hip_bf16.h>
#include <math.h>

#define NN 512
#define TCOLS 32            // tile width (fp32 elems)
#define TILES_PER_RG 16     // 512 / TCOLS
#define RGS 4               // 16-row groups per wave (8 waves * 64 rows = 512)
#define NTILES (RGS * TILES_PER_RG)

typedef __attribute__((ext_vector_type(2))) float v2f;
typedef __attribute__((ext_vector_type(8))) float v8f;
typedef __attribute__((ext_vector_type(4))) int   v4i;

#if defined(__gfx1250__) && \
    __has_builtin(__builtin_amdgcn_global_load_async_to_lds_b128) && \
    __has_builtin(__builtin_amdgcn_s_wait_asynccnt)
#define USE_ASYNC 1
typedef __attribute__((address_space(1))) v4i g_v4i;  // global
typedef __attribute__((address_space(3))) v4i l_v4i;  // LDS
#else
#define USE_ASYNC 0
#endif

// ---------------------------------------------------------------------------
// Stage 1: fold the DCT basis into the rank-1 weights.
//   u[k] = sum_i w_h[i] * cu[i] * cos((2k+1) * i * pi / (2N))   (v likewise)
// Integer angle reduced mod 4N (=2048) so cosf always sees arg < 2*pi.
// ---------------------------------------------------------------------------
__global__ __launch_bounds__(512) void dct_uv_kernel(
    const float* __restrict__ wh, const float* __restrict__ wv,
    float* __restrict__ uv) {
  const int k = threadIdx.x;              // 0..511
  const float step = 3.06796157577e-3f;   // pi / 1024
  const float cu0 = 0.04419417382f;       // sqrt(1/512)
  const float cuN = 0.0625f;              // sqrt(2/512) = 1/16
  float su = 0.f, sv = 0.f;
  const int tk = 2 * k + 1;
  for (int i = 0; i < NN; ++i) {
    int m = (tk * i) & 2047;              // exact mod 2*pi
    float bc = ((i == 0) ? cu0 : cuN) * cosf((float)m * step);
    su += wh[i] * bc;
    sv += wv[i] * bc;
  }
  uv[k]      = su;
  uv[NN + k] = sv;
}

// ---------------------------------------------------------------------------
// Stage 2: out[b] = sigmoid( u^T * X_b * v + bias ), X_b is 512x512 fp32.
// One workgroup (8 wave32) per batch; each wave independently owns 64 rows.
// Per-wave double-buffered 16x32 LDS tiles filled by async global->LDS b128
// (ASYNCcnt pipeline), consumed by V_WMMA_F32_16X16X4_F32 with
// B[k,n] = v[l+k] broadcast across n. No block barriers in the main loop.
// ---------------------------------------------------------------------------
__global__ __launch_bounds__(256) void dct_reduce_kernel(
    const float* __restrict__ x, const float* __restrict__ uv,
    const float* __restrict__ bias, float* __restrict__ out) {
  __shared__ float u_s[NN];                               // 2 KB
  __shared__ float v_s[NN];                               // 2 KB
  __shared__ __align__(16) float tiles[8][2][16 * TCOLS]; // 32 KB
  __shared__ float red[8];

  const int b    = blockIdx.x;
  const int tid  = threadIdx.x;
  const int wave = tid >> 5;
  const int lane = tid & 31;
  const int lm   = lane & 15;   // A: M row / C,D: N col
  const int hi   = lane >> 4;   // A,B: K pair {0,1} vs {2,3}; C,D: M +8

  for (int i = tid; i < NN; i += 256) {
    u_s[i] = uv[i];
    v_s[i] = uv[NN + i];
  }
  if (tid < 8) red[tid] = 0.f;
  __syncthreads();

  const float* xb = x + (size_t)b * (size_t)(NN * NN);
  float (*buf)[16 * TCOLS] = tiles[wave];

  // async/fallback tile prefetch: 16 rows x 32 cols, 4 x b128 per lane
  auto prefetch = [&](int t) {
    const int rg = t >> 4;
    const int l0 = (t & 15) * TCOLS;
    const int k0 = wave * 64 + rg * 16;
    const float* src = xb + (size_t)k0 * NN + l0;
    float* dst = buf[t & 1];
    const int row_off = (lane >> 3);      // 0..3
    const int col     = (lane & 7) * 4;   // 0..28
#pragma unroll
    for (int i = 0; i < 4; ++i) {
      const int row = i * 4 + row_off;
#if USE_ASYNC
      __builtin_amdgcn_global_load_async_to_lds_b128(
          (g_v4i*)(src + row * NN + col),
          (l_v4i*)(dst + row * TCOLS + col),
          0, 0);
#else
      *(float4*)(dst + row * TCOLS + col) =
          *(const float4*)(src + row * NN + col);
#endif
    }
  };

  float wave_total = 0.f;
  v8f acc = {0.f, 0.f, 0.f, 0.f, 0.f, 0.f, 0.f, 0.f};

  prefetch(0);
  for (int t = 0; t < NTILES; ++t) {
    if (t + 1 < NTILES) {
      prefetch(t + 1);                      // overlap next tile with compute
#if USE_ASYNC
      __builtin_amdgcn_s_wait_asynccnt(4);  // in-order: tile t has landed
#endif
    } else {
#if USE_ASYNC
      __builtin_amdgcn_s_wait_asynccnt(0);
#endif
    }
    asm volatile("" ::: "memory");          // keep LDS reads below the wait

    const float* tb = buf[t & 1];
    const int l0 = (t & 15) * TCOLS;
#pragma unroll
    for (int c0 = 0; c0 < TCOLS; c0 += 4) {
      // A[m,k] = X[k0+m, l0+c0+k]  (ISA 16x4 fp32 A layout)
      v2f a = *(const v2f*)(tb + lm * TCOLS + c0 + 2 * hi);
      // B[k,n] = v[l0+c0+k], broadcast across n
      const int lidx = l0 + c0 + 2 * hi;
      v2f bv;
      bv.x = v_s[lidx];
      bv.y = v_s[lidx + 1];
      acc = __builtin_amdgcn_wmma_f32_16x16x4_f32(
          false, a, false, bv, (short)0, acc, false, false);
    }

    if ((t & 15) == 15) {                   // end of a 16-row group: reduce
      const int rg = t >> 4;
      const int k0 = wave * 64 + rg * 16;
      const int base = k0 + 8 * hi;
      float p = 0.f;
#pragma unroll
      for (int g = 0; g < 8; ++g) p += u_s[base + g] * acc[g];
      wave_total += __shfl(p, 0) + __shfl(p, 16);
      acc = (v8f){0.f, 0.f, 0.f, 0.f, 0.f, 0.f, 0.f, 0.f};
    }
  }

  if (lane == 0) red[wave] = wave_total;
  __syncthreads();
  if (tid == 0) {
    float t = bias[0];
#pragma unroll
    for (int w = 0; w < 8; ++w) t += red[w];
    out[b] = 1.0f / (1.0f + expf(-t));
  }
}

// ---------------------------------------------------------------------------
extern "C" void kernel_launch(void* const* d_in, const int* in_sizes, int n_in,
                              void* d_out, int out_size, void* d_ws, size_t ws_size,
                              hipStream_t stream) {
  const float* x    = (const float*)d_in[0];
  const float* wh   = (const float*)d_in[1];
  const float* wv   = (const float*)d_in[2];
  const float* bias = (const float*)d_in[3];
  float* out = (float*)d_out;
  float* uv  = (float*)d_ws;                 // 2 * 512 floats = 4 KB scratch

  const int batch = in_sizes[0] / (NN * NN); // 256

  dct_uv_kernel<<<1, NN, 0, stream>>>(wh, wv, uv);
  dct_reduce_kernel<<<batch, 256, 0, stream>>>(x, uv, bias, out);
}